// LombScargle_1649267441797
// MI455X (gfx1250) — compile-verified
//
#include <hip/hip_runtime.h>
#include <math.h>

typedef float v2f __attribute__((ext_vector_type(2)));
typedef float v8f __attribute__((ext_vector_type(8)));

#define B_  16
#define N_  1024
#define F_  2048
#define FTILES (F_/16)          // 128 frequency tiles of 16
#define WAVES_PER_BLOCK 8       // 256 threads = 8 wave32

// ---------------------------------------------------------------------------
// Stage 1: one wave per (batch, 16-frequency tile).
// Reductions over N done with V_WMMA_F32_16X16X4_F32 on the matrix pipe:
//   A (16x4, f32): rows = frequencies, K interleaved as
//       K=2i   -> sin(w t_i)  (or cos)            [wt terms]
//       K=2i+1 -> sin(2w t_i) (or cos)            [2wt terms, double-angle]
//   B (4x16, f32): row 2i, col0 = m*y at sample i ; row 2i+1, col1 = m.
//   => D col0 = Sum m*y*trig(wt), D col1 = Sum m*trig(2wt).
// Lane mapping matches the ISA A-layout exactly (lanes 0-15: K=0,1;
// lanes 16-31: K=2,3), so sincos results land in-place, no shuffles.
// ---------------------------------------------------------------------------
__global__ __launch_bounds__(256) void ls_stage1(
    const float* __restrict__ t, const float* __restrict__ y,
    const float* __restrict__ mask, const float* __restrict__ freqs,
    float* __restrict__ p_out)
{
  __shared__ float xch[WAVES_PER_BLOCK][4][16];   // YS,S2,YC,C2 per wave

  const int lane  = threadIdx.x & 31;
  const int wave  = threadIdx.x >> 5;
  const int wid   = blockIdx.x * WAVES_PER_BLOCK + wave;
  const int b     = wid / FTILES;
  const int ftile = wid % FTILES;
  const int frow  = lane & 15;   // frequency row (M) / B column
  const int khalf = lane >> 4;   // 0: K=0,1 (sample n) ; 1: K=2,3 (sample n+1)
  const int f_idx = ftile * 16 + frow;
  const float f   = freqs[f_idx];        // cycles/unit; omega = 2*pi*f

  const float* __restrict__ tb = t    + b * N_;
  const float* __restrict__ yb = y    + b * N_;
  const float* __restrict__ mb = mask + b * N_;

  v8f acc_s = {0.f,0.f,0.f,0.f,0.f,0.f,0.f,0.f};
  v8f acc_c = {0.f,0.f,0.f,0.f,0.f,0.f,0.f,0.f};
  float mm = 0.f;                        // Sum m^2 (per n-parity half)

  for (int n = 0; n < N_; n += 2) {
    const int n_me = n + khalf;
    const float tv = tb[n_me];
    const float yv = yb[n_me];
    const float mv = mb[n_me];

    // phase in revolutions: sin(2*pi*f*t) == v_sin_f32(f*t)
    float rev  = f * tv;
    float frac = rev - __builtin_rintf(rev);     // range-reduce to [-0.5,0.5]
    float s = __builtin_amdgcn_sinf(frac);
    float c = __builtin_amdgcn_cosf(frac);
    float s2 = 2.0f * s * c;                         // sin(2wt)
    float c2 = __builtin_fmaf(-2.0f * s, s, 1.0f);   // cos(2wt)

    v2f a_s = { s, s2 };
    v2f a_c = { c, c2 };

    float my = mv * yv;
    v2f bm;
    bm.x = (frow == 0) ? my : 0.0f;   // rows K0|K2: col 0 <- m*y
    bm.y = (frow == 1) ? mv : 0.0f;   // rows K1|K3: col 1 <- m

    acc_s = __builtin_amdgcn_wmma_f32_16x16x4_f32(
        false, a_s, false, bm, (short)0, acc_s, false, false);
    acc_c = __builtin_amdgcn_wmma_f32_16x16x4_f32(
        false, a_c, false, bm, (short)0, acc_c, false, false);

    mm = __builtin_fmaf(mv, mv, mm);
  }
  // combine the two n-parity halves of Sum m^2
  mm += __shfl_xor(mm, 16);

  // Spill D columns 0 (Y-weighted) and 1 (mask-weighted) to LDS.
  // D layout: lane = column (+16 for M>=8 half), VGPR r -> M = khalf*8 + r.
  if (frow < 2) {
    const int mbase = khalf * 8;
#pragma unroll
    for (int r = 0; r < 8; ++r) {
      xch[wave][0 + frow][mbase + r] = acc_s[r];  // 0: YS, 1: S2
      xch[wave][2 + frow][mbase + r] = acc_c[r];  // 2: YC, 3: C2
    }
  }
  __syncthreads();

  if (lane < 16) {
    const float YS = xch[wave][0][lane];
    const float S2 = xch[wave][1][lane];
    const float YC = xch[wave][2][lane];
    const float C2 = xch[wave][3][lane];

    // 2*w*tau = atan2(S2, C2)  =>  w*tau in revolutions = atan2 / (4*pi)
    float ph = atan2f(S2, C2) * 0.07957747154594767f;  // 1/(4*pi)
    float ct = __builtin_amdgcn_cosf(ph);
    float st = __builtin_amdgcn_sinf(ph);

    float cc = ct * ct, ss = st * st, cs = ct * st;
    // Sum cos_term^2 and Sum sin_term^2 via angle addition
    float Ecc = 0.5f * (cc * (mm + C2) + ss * (mm - C2)) + cs * S2;
    float Ess = 0.5f * (cc * (mm - C2) + ss * (mm + C2)) - cs * S2;
    float yct = ct * YC + st * YS;   // Sum y*cos_term
    float yst = ct * YS - st * YC;   // Sum y*sin_term

    float p = 0.5f * (yct * yct / (Ecc + 1e-10f) +
                      yst * yst / (Ess + 1e-10f));

    // false-alarm-probability correction: fap = 1 - (1 - e^-p)^F
    float base1 = 1.0f - __expf(-p);
    float fap   = 1.0f - __powf(base1, (float)F_);
    p = p / (fap + 1e-5f);

    p_out[b * F_ + ftile * 16 + lane] = p;
  }
}

// ---------------------------------------------------------------------------
// Stage 2: per-batch trapezoid integral over F, in-place normalization.
// ---------------------------------------------------------------------------
__global__ __launch_bounds__(256) void ls_stage2(
    const float* __restrict__ freqs, float* __restrict__ p)
{
  __shared__ float red[256];
  const int b   = blockIdx.x;
  const int tid = threadIdx.x;
  float* __restrict__ pb = p + b * F_;

  float partial = 0.f;
  for (int fi = tid; fi < F_ - 1; fi += 256) {
    float df = freqs[fi + 1] - freqs[fi];
    partial += 0.5f * (pb[fi] + pb[fi + 1]) * df;
  }
  red[tid] = partial;
  __syncthreads();
  for (int s = 128; s > 0; s >>= 1) {
    if (tid < s) red[tid] += red[tid + s];
    __syncthreads();
  }
  const float inv = 1.0f / (red[0] + 1e-10f);
  // all phase-1 reads are complete (final barrier above); each thread
  // rewrites only its own elements -> no races, deterministic.
  for (int fi = tid; fi < F_; fi += 256) {
    pb[fi] *= inv;
  }
}

extern "C" void kernel_launch(void* const* d_in, const int* in_sizes, int n_in,
                              void* d_out, int out_size, void* d_ws, size_t ws_size,
                              hipStream_t stream) {
  (void)in_sizes; (void)n_in; (void)d_ws; (void)ws_size; (void)out_size;
  const float* t     = (const float*)d_in[0];
  const float* y     = (const float*)d_in[1];
  const float* mask  = (const float*)d_in[2];
  const float* freqs = (const float*)d_in[3];
  float* out = (float*)d_out;

  const int tiles  = B_ * FTILES;                 // 2048 waves
  const int blocks = tiles / WAVES_PER_BLOCK;     // 256 blocks of 8 waves
  ls_stage1<<<blocks, 256, 0, stream>>>(t, y, mask, freqs, out);
  ls_stage2<<<B_, 256, 0, stream>>>(freqs, out);
}